// CCLoss_14663018349147
// MI455X (gfx1250) — compile-verified
//
#include <hip/hip_runtime.h>
#include <hip/hip_bf16.h>
#include <math.h>

typedef float v2f __attribute__((ext_vector_type(2)));
typedef float v8f __attribute__((ext_vector_type(8)));

#define N_PER_IMG 307200      // 1*480*640
#define BATCH     64
#define CHUNKS    75          // blocks per image in pass 1 (75*4096 = 307200)
#define SLOTS     80          // padded to multiple of 4 -> 20 WMMA K-groups
#define QUANT     5           // Sx, Sy, Sxx, Syy, Sxy
#define CC_EPS    1e-8f

// ---------------- init: zero the partial-sum workspace (incl. padding) ----
__global__ void cc_zero_ws(float* __restrict__ p, int n) {
    int i = blockIdx.x * blockDim.x + threadIdx.x;
    if (i < n) p[i] = 0.0f;
}

// ---------------- pass 1: streaming partial sums (bandwidth-bound) -------
__global__ __launch_bounds__(256)
void cc_partials(const float* __restrict__ x, const float* __restrict__ y,
                 float* __restrict__ P) {
    const int img   = blockIdx.x / CHUNKS;
    const int chunk = blockIdx.x % CHUNKS;
    const size_t base = (size_t)img * N_PER_IMG + (size_t)chunk * 4096;
    const float4* __restrict__ xv = (const float4*)(x + base);
    const float4* __restrict__ yv = (const float4*)(y + base);

    float sx = 0.f, sy = 0.f, sxx = 0.f, syy = 0.f, sxy = 0.f;
#pragma unroll
    for (int k = 0; k < 4; ++k) {
        float4 a = xv[threadIdx.x + 256 * k];
        float4 b = yv[threadIdx.x + 256 * k];
        sx  += a.x + a.y + a.z + a.w;
        sy  += b.x + b.y + b.z + b.w;
        sxx += a.x*a.x + a.y*a.y + a.z*a.z + a.w*a.w;
        syy += b.x*b.x + b.y*b.y + b.z*b.z + b.w*b.w;
        sxy += a.x*b.x + a.y*b.y + a.z*b.z + a.w*b.w;
    }

    // wave32 reduction
#pragma unroll
    for (int off = 16; off > 0; off >>= 1) {
        sx  += __shfl_down(sx,  off);
        sy  += __shfl_down(sy,  off);
        sxx += __shfl_down(sxx, off);
        syy += __shfl_down(syy, off);
        sxy += __shfl_down(sxy, off);
    }

    __shared__ float red[8][QUANT];
    const int wave = threadIdx.x >> 5;
    const int lane = threadIdx.x & 31;
    if (lane == 0) {
        red[wave][0] = sx;  red[wave][1] = sy;  red[wave][2] = sxx;
        red[wave][3] = syy; red[wave][4] = sxy;
    }
    __syncthreads();
    if (threadIdx.x == 0) {
        float t[QUANT] = {0.f, 0.f, 0.f, 0.f, 0.f};
#pragma unroll
        for (int w = 0; w < 8; ++w)
#pragma unroll
            for (int q = 0; q < QUANT; ++q) t[q] += red[w][q];
#pragma unroll
        for (int q = 0; q < QUANT; ++q)
            P[q * (BATCH * SLOTS) + img * SLOTS + chunk] = t[q];
    }
}

// ---------------- pass 2: WMMA ones-matrix reduction + closed-form r -----
// Each of 20 waves owns (quantity q, 16-image group): D += A(16x4) * ones(4x16)
// chained 20x contracts all 80 slots exactly in the f32 accumulator.
__global__ __launch_bounds__(640)
void cc_finalize(const float* __restrict__ P, float* __restrict__ out) {
    __shared__ float sums[QUANT][BATCH];
    __shared__ float rbuf[BATCH];

    const int tid  = threadIdx.x;
    const int wave = tid >> 5;          // 0..19
    const int lane = tid & 31;
    const int q    = wave / 4;          // quantity 0..4
    const int grp  = wave % 4;          // image group of 16
    const int m    = lane & 15;         // A-matrix row (image within group)
    const int k0   = (lane < 16) ? 0 : 2;  // ISA A-layout: lanes 16-31 hold K=2,3

    const float* __restrict__ row =
        P + q * (BATCH * SLOTS) + (size_t)(grp * 16 + m) * SLOTS;

    v8f c = {};
    v2f b; b.x = 1.0f; b.y = 1.0f;      // B = ones(4x16) -> D[m][n] = row-sum(A)
#pragma unroll
    for (int g = 0; g < 20; ++g) {
        v2f a;
        a.x = row[4 * g + k0];
        a.y = row[4 * g + k0 + 1];
        c = __builtin_amdgcn_wmma_f32_16x16x4_f32(
                /*neg_a=*/false, a, /*neg_b=*/false, b,
                /*c_mod=*/(short)0, c, /*reuse_a=*/false, /*reuse_b=*/false);
    }

    // D layout: VGPR j = rows {j, j+8}; N = lane%16. Lanes with N==0 extract.
    if (m == 0) {
        const int ibase = grp * 16 + ((lane == 16) ? 8 : 0);
#pragma unroll
        for (int j = 0; j < 8; ++j) sums[q][ibase + j] = c[j];
    }
    __syncthreads();

    if (tid < BATCH) {
        const float n  = (float)N_PER_IMG;
        const float Sx  = sums[0][tid], Sy  = sums[1][tid];
        const float Sxx = sums[2][tid], Syy = sums[3][tid];
        const float Sxy = sums[4][tid];

        const float cxx = Sxx - Sx * Sx / n;   // centered sum of squares
        const float cyy = Syy - Sy * Sy / n;
        const float cxy = Sxy - Sx * Sy / n;

        // torch.std: unbiased (ddof=1); reference adds EPS to sd
        const float sda = sqrtf(fmaxf(cxx / (n - 1.0f), 0.f)) + CC_EPS;
        const float sdb = sqrtf(fmaxf(cyy / (n - 1.0f), 0.f)) + CC_EPS;
        const float ca = 1.0f / sda, cb = 1.0f / sdb;

        const float num = ca * cb * cxy;               // sum(a*b)
        const float sa2 = ca * ca * cxx;               // sum(a*a)
        const float sb2 = cb * cb * cyy;               // sum(b*b)
        rbuf[tid] = num / sqrtf(sa2 * sb2 + CC_EPS);
    }
    __syncthreads();

    if (tid == 0) {
        float s = 0.f;
#pragma unroll
        for (int i = 0; i < BATCH; ++i) s += rbuf[i];
        out[0] = -s / (float)BATCH;
    }
}

extern "C" void kernel_launch(void* const* d_in, const int* in_sizes, int n_in,
                              void* d_out, int out_size, void* d_ws, size_t ws_size,
                              hipStream_t stream) {
    (void)in_sizes; (void)n_in; (void)out_size; (void)ws_size;
    const float* y_pred = (const float*)d_in[0];
    const float* y_true = (const float*)d_in[1];
    float* out = (float*)d_out;
    float* P   = (float*)d_ws;          // [QUANT][BATCH][SLOTS] f32 = 100 KB

    const int nP = QUANT * BATCH * SLOTS;
    cc_zero_ws<<<(nP + 255) / 256, 256, 0, stream>>>(P, nP);
    cc_partials<<<BATCH * CHUNKS, 256, 0, stream>>>(y_pred, y_true, P);
    cc_finalize<<<1, 640, 0, stream>>>(P, out);
}